// ContrastiveLoss_matrix_26852135535215
// MI455X (gfx1250) — compile-verified
//
#include <hip/hip_runtime.h>
#include <stdint.h>

typedef __attribute__((ext_vector_type(2))) float v2f;
typedef __attribute__((ext_vector_type(8))) float v8f;

#define NEG_INF     (-1e30f)
#define LOSS_EPS    (1e-6f)
#define LOSS_MARGIN (0.2f)

// CDNA5 async global->LDS copy (ASYNCcnt path, no VGPR round-trip).
// VDST = per-lane LDS byte offset, VADDR = 64-bit global address.
__device__ __forceinline__ void async_load_b128(float* lds_dst, const float* gsrc) {
    uint32_t lds_off = (uint32_t)(uintptr_t)lds_dst;  // low 32 bits of flat LDS addr = LDS byte offset
    asm volatile("global_load_async_to_lds_b128 %0, %1, off"
                 :: "v"(lds_off), "v"(gsrc)
                 : "memory");
}
__device__ __forceinline__ void wait_async_0() {
    asm volatile("s_wait_asynccnt 0x0" ::: "memory");
}

// 512-thread block (16 waves) covers a 4-image x 4-caption tile; wave w computes
// pair (bi = w>>2, cj = w&3). A/B K-slices are staged in LDS via async B128 copies
// (pad rows zeroed once -> branch-free WMMA loop), fragments fed by ds_load_b64,
// math done with exact-fp32 V_WMMA_F32_16X16X4_F32.
template<int R, int W, int RT, int WT, bool ACCUM>
__global__ __launch_bounds__(512)
void xattn_i2t_kernel(const float* __restrict__ img,
                      const int*   __restrict__ img_len,
                      const float* __restrict__ cap,
                      const int*   __restrict__ cap_len,
                      float* __restrict__ scores,
                      int C, int K)
{
    constexpr int RPAD = RT * 16;
    constexpr int WPAD = WT * 16;
    constexpr int KC   = 32;        // K-chunk per staging round
    constexpr int QA   = KC / 4;    // B128 quads per row-chunk
    constexpr int KS   = KC + 4;    // LDS row stride (floats): 16B-aligned, bank-conflict-free

    extern __shared__ float smem[];
    float* AS = smem;                     // [4][RPAD][KS] image panels
    float* BS = smem + 4 * RPAD * KS;     // [4][WPAD][KS] caption panels

    const int tid  = threadIdx.x;
    const int wid  = tid >> 5;
    const int lane = tid & 31;
    const int half = lane >> 4;           // K pair selector within 16x16x4 step
    const int ln   = lane & 15;
    const int bi   = wid >> 2;
    const int cj   = wid & 3;

    const int b0 = (blockIdx.x / (C / 4)) * 4;
    const int c0 = (blockIdx.x % (C / 4)) * 4;

    // Zero the LDS pad rows once (A rows >= R, B rows >= W). Never rewritten.
    for (int i = tid; i < 4 * (RPAD - R) * KS; i += 512) {
        const int p   = i / ((RPAD - R) * KS);
        const int rem = i - p * ((RPAD - R) * KS);
        AS[(p * RPAD + R) * KS + rem] = 0.f;
    }
    for (int i = tid; i < 4 * (WPAD - W) * KS; i += 512) {
        const int p   = i / ((WPAD - W) * KS);
        const int rem = i - p * ((WPAD - W) * KS);
        BS[(p * WPAD + W) * KS + rem] = 0.f;
    }

    v8f acc[RT][WT];
#pragma unroll
    for (int rt = 0; rt < RT; ++rt)
#pragma unroll
        for (int wt = 0; wt < WT; ++wt)
            acc[rt][wt] = (v8f){0.f,0.f,0.f,0.f,0.f,0.f,0.f,0.f};

    for (int kk = 0; kk < K; kk += KC) {
        // Stage A: 4 images x R rows x QA quads (B128 each), async to LDS.
        for (int i = tid; i < 4 * R * QA; i += 512) {
            const int p   = i / (R * QA);
            const int rem = i - p * (R * QA);
            const int row = rem / QA;
            const int q   = rem - row * QA;
            async_load_b128(&AS[(p * RPAD + row) * KS + q * 4],
                            img + ((size_t)(b0 + p) * R + row) * K + kk + q * 4);
        }
        // Stage B: 4 captions x W rows x QA quads.
        for (int i = tid; i < 4 * W * QA; i += 512) {
            const int p   = i / (W * QA);
            const int rem = i - p * (W * QA);
            const int row = rem / QA;
            const int q   = rem - row * QA;
            async_load_b128(&BS[(p * WPAD + row) * KS + q * 4],
                            cap + ((size_t)(c0 + p) * W + row) * K + kk + q * 4);
        }
        wait_async_0();          // this wave's async copies done
        __syncthreads();         // all waves' copies visible

        const float* Ap = AS + bi * RPAD * KS;
        const float* Bp = BS + cj * WPAD * KS;
#pragma unroll
        for (int ks = 0; ks < KC / 4; ++ks) {
            const int k2 = ks * 4 + half * 2;
            v2f af[RT];
            v2f bf[WT];
#pragma unroll
            for (int rt = 0; rt < RT; ++rt)
                af[rt] = *(const v2f*)(Ap + (rt * 16 + ln) * KS + k2);
#pragma unroll
            for (int wt = 0; wt < WT; ++wt)
                bf[wt] = *(const v2f*)(Bp + (wt * 16 + ln) * KS + k2);
#pragma unroll
            for (int rt = 0; rt < RT; ++rt)
#pragma unroll
                for (int wt = 0; wt < WT; ++wt)
                    acc[rt][wt] = __builtin_amdgcn_wmma_f32_16x16x4_f32(
                        false, af[rt], false, bf[wt],
                        (short)0, acc[rt][wt], false, false);
        }
        __syncthreads();         // all waves done reading before next-chunk overwrite
    }

    // Epilogue. C/D layout: element (m,n) lives in lane (m>=8)*16 + n, VGPR m&7.
    const int b = b0 + bi;
    const int c = c0 + cj;
    const int nwords = cap_len[c];
    float region_sum = 0.f;
#pragma unroll
    for (int rt = 0; rt < RT; ++rt) {
#pragma unroll
        for (int j = 0; j < 8; ++j) {
            float mx = NEG_INF;
#pragma unroll
            for (int wt = 0; wt < WT; ++wt) {
                const int w = wt * 16 + ln;
                const float v = acc[rt][wt][j];
                mx = fmaxf(mx, (w < nwords) ? v : NEG_INF);
            }
            mx = fmaxf(mx, __shfl_xor(mx, 1, 32));
            mx = fmaxf(mx, __shfl_xor(mx, 2, 32));
            mx = fmaxf(mx, __shfl_xor(mx, 4, 32));
            mx = fmaxf(mx, __shfl_xor(mx, 8, 32));
            region_sum += mx;    // row m = rt*16 + half*8 + j (zero-pad rows add 0)
        }
    }
    const float total = __shfl(region_sum, 0, 32) + __shfl(region_sum, 16, 32);
    if (lane == 0) {
        const float sim = total / ((float)img_len[b] + LOSS_EPS);
        const size_t idx = (size_t)b * C + c;
        if (ACCUM) scores[idx] = scores[idx] + sim;   // PRED_RATE == 1.0
        else       scores[idx] = sim;
    }
}

// Hinge-loss reduction over the BxB score matrix. Single block.
__global__ __launch_bounds__(256)
void hinge_loss_kernel(const float* __restrict__ scores,
                       float* __restrict__ loss_out, int B)
{
    __shared__ float diag[256];
    __shared__ float wsum[8];
    const int tid = threadIdx.x;

    for (int i = tid; i < B; i += 256) diag[i] = scores[(size_t)i * B + i];
    __syncthreads();

    float acc = 0.f;
    const int N = B * B;
    for (int idx = tid; idx < N; idx += 256) {
        const int i = idx / B;
        const int j = idx - i * B;
        if (i != j) {
            const float sc = scores[idx];
            acc += fmaxf(0.f, LOSS_MARGIN + sc - diag[i]);  // cost_s  (row diag)
            acc += fmaxf(0.f, LOSS_MARGIN + sc - diag[j]);  // cost_im (col diag)
        }
    }
    acc += __shfl_xor(acc, 16, 32);
    acc += __shfl_xor(acc, 8, 32);
    acc += __shfl_xor(acc, 4, 32);
    acc += __shfl_xor(acc, 2, 32);
    acc += __shfl_xor(acc, 1, 32);
    if ((tid & 31) == 0) wsum[tid >> 5] = acc;
    __syncthreads();
    if (tid == 0) {
        float t = 0.f;
        for (int w = 0; w < 8; ++w) t += wsum[w];
        loss_out[0] = t;
    }
}

extern "C" void kernel_launch(void* const* d_in, const int* in_sizes, int n_in,
                              void* d_out, int out_size, void* d_ws, size_t ws_size,
                              hipStream_t stream) {
    const float* im       = (const float*)d_in[0];
    const int*   im_l     = (const int*)  d_in[1];
    const float* s        = (const float*)d_in[2];
    const int*   s_l      = (const int*)  d_in[3];
    const float* pred     = (const float*)d_in[4];
    const int*   pred_l   = (const int*)  d_in[5];
    const float* s_pred   = (const float*)d_in[6];
    const int*   s_pred_l = (const int*)  d_in[7];

    float* out    = (float*)d_out;
    float* scores = out + 1;                 // [loss, scores(BxB)] layout

    const int B = in_sizes[1];               // 128
    const int K = in_sizes[0] / (B * 36);    // 1024
    const int nblk = (B / 4) * (B / 4);      // 4x4 (image,caption) tiles per block

    constexpr int KS = 36;                   // LDS row stride in floats
    const size_t smem1 = (size_t)(4 * 48 + 4 * 64) * KS * sizeof(float);  // 64,512 B
    const size_t smem2 = (size_t)(4 * 32 + 4 * 32) * KS * sizeof(float);  // 36,864 B

    // Pass 1: R=36 (3 row tiles), W=50 (4 col tiles)
    xattn_i2t_kernel<36, 50, 3, 4, false>
        <<<dim3(nblk), dim3(512), smem1, stream>>>(im, im_l, s, s_l, scores, B, K);
    // Pass 2: R=25 (2 row tiles), W=30 (2 col tiles), accumulate (PRED_RATE=1)
    xattn_i2t_kernel<25, 30, 2, 2, true>
        <<<dim3(nblk), dim3(512), smem2, stream>>>(pred, pred_l, s_pred, s_pred_l, scores, B, K);
    // Loss reduction
    hinge_loss_kernel<<<dim3(1), dim3(256), 0, stream>>>(scores, out, B);
}